// Property_prediction_deep_13116830122573
// MI455X (gfx1250) — compile-verified
//
#include <hip/hip_runtime.h>
#include <hip/hip_bf16.h>
#include <math.h>

#define N_ATOM 50000
#define M_NBR  12
#define NBRF   41
#define AF     64
#define ORIG   92
#define NCONV  3
#define NCRY   500
#define NA     100
#define NM     (N_ATOM * M_NBR)        // 600000
#define KTOT   (2 * AF + NBRF)         // 169
#define KPAD   172                     // pad to multiple of 4
#define OC     (2 * AF)                // 128
#define BN_EPS 1e-5f

typedef __attribute__((ext_vector_type(2))) float v2f;
typedef __attribute__((ext_vector_type(8))) float v8f;

__device__ __forceinline__ float sp(float x) {
    // stable softplus: max(x,0) + log1p(exp(-|x|))
    return fmaxf(x, 0.f) + log1pf(expf(-fabsf(x)));
}
__device__ __forceinline__ float sigm(float x) {
    return 1.f / (1.f + expf(-x));
}

// out[500 + i] = atom_fea[i] * mask[i % 92]
__global__ __launch_bounds__(256) void amf_kernel(const float* __restrict__ atom_fea,
                                                  const float* __restrict__ mask,
                                                  float* __restrict__ out_amf) {
    int i = blockIdx.x * blockDim.x + threadIdx.x;
    if (i < N_ATOM * ORIG) out_amf[i] = atom_fea[i] * mask[i % ORIG];
}

// h0[n,c] = sum_k atom_fea[n,k]*mask[k]*emb_W[c,k]
__global__ __launch_bounds__(256) void embed_kernel(const float* __restrict__ atom_fea,
                                                    const float* __restrict__ mask,
                                                    const float* __restrict__ emb_W,
                                                    float* __restrict__ h0) {
    int i = blockIdx.x * blockDim.x + threadIdx.x;
    if (i >= N_ATOM * AF) return;
    int n = i >> 6, c = i & 63;
    const float* a = atom_fea + n * ORIG;
    const float* w = emb_W + c * ORIG;
    float s = 0.f;
    for (int k = 0; k < ORIG; ++k) s += a[k] * mask[k] * w[k];
    h0[i] = s;
}

// Wpad[l][c][0..171] = conv_W[l][c][k] for k<169 else 0   (3*128*172 elems)
__global__ __launch_bounds__(256) void pad_w_kernel(const float* __restrict__ conv_W,
                                                    float* __restrict__ Wpad) {
    int i = blockIdx.x * blockDim.x + threadIdx.x;
    if (i >= NCONV * OC * KPAD) return;
    int k = i % KPAD;
    int lc = i / KPAD;                       // layer*128 + channel
    Wpad[i] = (k < KTOT) ? conv_W[lc * KTOT + k] : 0.f;
}

__global__ void zero_stats_kernel(float* __restrict__ stats1, float* __restrict__ stats2) {
    int t = threadIdx.x;
    if (t < 256) stats1[t] = 0.f;
    else if (t < 384) stats2[t - 256] = 0.f;
}

// gated(NM x 128) = [h | h[nbr] | nbr_fea] (NM x 169pad172) @ Wpad^T + bW,
// accumulating per-channel sum/sumsq into stats1 ([sum(128) | sumsq(128)]).
// One block = 32 rows; 8 waves each own 16 output channels and two 16x16
// C tiles (rows 0-15 and 16-31), sharing the same B fragment per K step.
// Inner loop: 2x V_WMMA_F32_16X16X4_F32, no branches, no EXEC masking.
__global__ __launch_bounds__(256) void conv_gemm_kernel(const float* __restrict__ h,
                                                        const float* __restrict__ nbr_fea,
                                                        const int* __restrict__ nbr_idx,
                                                        const float* __restrict__ Wpad,  // 128 x 172
                                                        const float* __restrict__ bW,    // 128
                                                        float* __restrict__ gated,       // NM x 128
                                                        float* __restrict__ stats1) {
    __shared__ float sT[32 * KPAD];   // 32-row tile of "total" features (zero-padded K)
    __shared__ float sred[2 * OC];    // block-level per-channel sum / sumsq

    const int tid  = threadIdx.x;
    const int row0 = blockIdx.x * 32;   // NM % 32 == 0, no tail

    sred[tid] = 0.f;

    for (int idx = tid; idx < 32 * KPAD; idx += 256) {
        int r = idx / KPAD, k = idx - r * KPAD;
        int row = row0 + r;
        int n = row / M_NBR;
        float v = 0.f;
        if (k < AF) {
            v = h[n * AF + k];
        } else if (k < 2 * AF) {
            v = h[nbr_idx[row] * AF + (k - AF)];
        } else if (k < KTOT) {
            v = nbr_fea[(size_t)row * NBRF + (k - 2 * AF)];
        }
        sT[idx] = v;
    }
    __syncthreads();

    const int wave = tid >> 5;          // 0..7 -> column tile
    const int lane = tid & 31;
    const int rowm = lane & 15;         // M within tile (A); N within tile (B/C)
    const int hi   = lane >> 4;         // 0: K={0,1}/M=j ; 1: K={2,3}/M=j+8
    const int col  = wave * 16 + rowm;  // output channel

    // aligned float2 views (K index always even -> 8B aligned)
    const v2f* wrow  = (const v2f*)(Wpad + (size_t)col * KPAD);
    const v2f* arow0 = (const v2f*)(sT + rowm * KPAD);
    const v2f* arow1 = (const v2f*)(sT + (16 + rowm) * KPAD);

    v8f acc0 = {}, acc1 = {};
    for (int k0 = 0; k0 < KPAD; k0 += 4) {
        int q = (k0 >> 1) + hi;         // float2 index: (k0 + 2*hi)/2
        v2f b  = wrow[q];
        v2f a0 = arow0[q];
        v2f a1 = arow1[q];
        acc0 = __builtin_amdgcn_wmma_f32_16x16x4_f32(false, a0, false, b,
                                                     (short)0, acc0, false, false);
        acc1 = __builtin_amdgcn_wmma_f32_16x16x4_f32(false, a1, false, b,
                                                     (short)0, acc1, false, false);
    }

    const float bias = bW[col];
    float ps = 0.f, pss = 0.f;
#pragma unroll
    for (int j = 0; j < 8; ++j) {
        // D layout: VGPR j -> M=j (lanes 0-15) / M=j+8 (lanes 16-31)
        float v0 = acc0[j] + bias;
        gated[(size_t)(row0 + j + 8 * hi) * OC + col] = v0;
        ps += v0; pss += v0 * v0;
        float v1 = acc1[j] + bias;
        gated[(size_t)(row0 + 16 + j + 8 * hi) * OC + col] = v1;
        ps += v1; pss += v1 * v1;
    }
    atomicAdd(&sred[col], ps);
    atomicAdd(&sred[OC + col], pss);
    __syncthreads();
    atomicAdd(&stats1[tid], sred[tid]);   // tid in [0,256): [sum | sumsq]
}

// coef1 = [a(128) | s(128)] with a=g*rsqrt(var+eps), s=b-mean*a  (denominator NM)
__global__ void bn1_final_kernel(const float* __restrict__ stats1,
                                 const float* __restrict__ g,
                                 const float* __restrict__ b,
                                 float* __restrict__ coef1) {
    int c = threadIdx.x;
    if (c >= OC) return;
    float inv_n = 1.f / (float)NM;
    float mean = stats1[c] * inv_n;
    float var  = stats1[OC + c] * inv_n - mean * mean;
    float a = g[c] * rsqrtf(var + BN_EPS);
    coef1[c] = a;
    coef1[OC + c] = b[c] - mean * a;
}

// summed[n,c] = sum_m sigmoid(BN(filt)) * softplus(BN(core)); one block per atom n
__global__ __launch_bounds__(64) void gate_reduce_kernel(const float* __restrict__ gated,
                                                         const float* __restrict__ coef1,
                                                         float* __restrict__ summed) {
    int n = blockIdx.x;
    int c = threadIdx.x;                 // 0..63
    const float* base = gated + (size_t)n * M_NBR * OC;
    float a_f = coef1[c],        s_f = coef1[OC + c];
    float a_c = coef1[AF + c],   s_c = coef1[OC + AF + c];
    float acc = 0.f;
#pragma unroll
    for (int m = 0; m < M_NBR; ++m) {
        float f  = base[m * OC + c]      * a_f + s_f;
        float co = base[m * OC + AF + c] * a_c + s_c;
        acc += sigm(f) * sp(co);
    }
    summed[n * AF + c] = acc;
}

// per-channel sum/sumsq of summed over N rows -> stats2 ([sum(64) | sumsq(64)])
__global__ __launch_bounds__(256) void stats2_kernel(const float* __restrict__ summed,
                                                     float* __restrict__ stats2) {
    __shared__ float s[AF], ss[AF];
    int t = threadIdx.x;
    if (t < AF) { s[t] = 0.f; ss[t] = 0.f; }
    __syncthreads();
    int c = t & 63;
    float ps = 0.f, pss = 0.f;
    for (int n = blockIdx.x * 4 + (t >> 6); n < N_ATOM; n += gridDim.x * 4) {
        float v = summed[n * AF + c];
        ps += v;
        pss += v * v;
    }
    atomicAdd(&s[c], ps);
    atomicAdd(&ss[c], pss);
    __syncthreads();
    if (t < AF) {
        atomicAdd(&stats2[t], s[t]);
        atomicAdd(&stats2[AF + t], ss[t]);
    }
}

__global__ void bn2_final_kernel(const float* __restrict__ stats2,
                                 const float* __restrict__ g,
                                 const float* __restrict__ b,
                                 float* __restrict__ coef2) {
    int c = threadIdx.x;
    if (c >= AF) return;
    float inv_n = 1.f / (float)N_ATOM;
    float mean = stats2[c] * inv_n;
    float var  = stats2[AF + c] * inv_n - mean * mean;
    float a = g[c] * rsqrtf(var + BN_EPS);
    coef2[c] = a;
    coef2[AF + c] = b[c] - mean * a;
}

// h_next = softplus(h + BN2(summed))
__global__ __launch_bounds__(256) void update_h_kernel(const float* __restrict__ h,
                                                       const float* __restrict__ summed,
                                                       const float* __restrict__ coef2,
                                                       float* __restrict__ h_next) {
    int i = blockIdx.x * blockDim.x + threadIdx.x;
    if (i >= N_ATOM * AF) return;
    int c = i & 63;
    float v = h[i] + summed[i] * coef2[c] + coef2[AF + c];
    h_next[i] = sp(v);
}

// per-crystal: L2-normalize gathered rows, mean-pool, MLP 64->64->64->1
__global__ __launch_bounds__(128) void readout_kernel(const float* __restrict__ h,
                                                      const int* __restrict__ cidx,
                                                      const float* __restrict__ fc1W,
                                                      const float* __restrict__ fc1b,
                                                      const float* __restrict__ fc2W,
                                                      const float* __restrict__ fc2b,
                                                      const float* __restrict__ outW,
                                                      const float* __restrict__ outb,
                                                      float* __restrict__ props) {
    __shared__ float inv[NA];
    __shared__ float pooled[AF];
    __shared__ float z1[AF];
    __shared__ float z2[AF];
    int b = blockIdx.x, t = threadIdx.x;
    const int* idx = cidx + b * NA;

    for (int a = t; a < NA; a += 128) {
        const float* row = h + (size_t)idx[a] * AF;
        float s = 0.f;
        for (int c = 0; c < AF; ++c) { float v = row[c]; s += v * v; }
        inv[a] = 1.f / fmaxf(sqrtf(s), 1e-12f);
    }
    __syncthreads();

    if (t < AF) {
        float s = 0.f;
        for (int a = 0; a < NA; ++a) s += h[(size_t)idx[a] * AF + t] * inv[a];
        pooled[t] = s * (1.f / (float)NA);
    }
    __syncthreads();

    if (t < AF) {
        float s = fc1b[t];
        for (int c = 0; c < AF; ++c) s += pooled[c] * fc1W[t * AF + c];
        z1[t] = sp(s);
    }
    __syncthreads();

    if (t < AF) {
        float s = fc2b[t];
        for (int c = 0; c < AF; ++c) s += z1[c] * fc2W[t * AF + c];
        z2[t] = sp(s);
    }
    __syncthreads();

    if (t == 0) {
        float s = outb[0];
        for (int c = 0; c < AF; ++c) s += z2[c] * outW[c];
        props[b] = s;
    }
}

extern "C" void kernel_launch(void* const* d_in, const int* in_sizes, int n_in,
                              void* d_out, int out_size, void* d_ws, size_t ws_size,
                              hipStream_t stream) {
    const float* atom_fea = (const float*)d_in[0];
    const float* nbr_fea  = (const float*)d_in[1];
    const int*   nbr_idx  = (const int*)d_in[2];
    const int*   cidx     = (const int*)d_in[3];
    const float* mask     = (const float*)d_in[4];
    const float* emb_W    = (const float*)d_in[5];
    const float* conv_W   = (const float*)d_in[6];   // 3 x 128 x 169
    const float* conv_b   = (const float*)d_in[7];   // 3 x 128
    const float* bn1_g    = (const float*)d_in[8];
    const float* bn1_b    = (const float*)d_in[9];
    const float* bn2_g    = (const float*)d_in[10];
    const float* bn2_b    = (const float*)d_in[11];
    const float* fc1_W    = (const float*)d_in[12];
    const float* fc1_b    = (const float*)d_in[13];
    const float* fc2_W    = (const float*)d_in[14];
    const float* fc2_b    = (const float*)d_in[15];
    const float* out_W    = (const float*)d_in[16];
    const float* out_b    = (const float*)d_in[17];

    float* props = (float*)d_out;                 // 500
    float* amf   = (float*)d_out + NCRY;          // 50000 x 92

    // workspace layout (floats)
    float* ws = (float*)d_ws;
    const size_t N64 = (size_t)N_ATOM * AF;
    float* hA     = ws;                        // N*64
    float* hB     = hA + N64;                  // N*64
    float* gated  = hB + N64;                  // NM*128
    float* summed = gated + (size_t)NM * OC;   // N*64
    float* stats1 = summed + N64;              // 256
    float* coef1  = stats1 + 256;              // 256
    float* stats2 = coef1 + 256;               // 128
    float* coef2  = stats2 + 128;              // 128
    float* Wpad   = coef2 + 128;               // 3*128*172

    // second output: masked atom features
    amf_kernel<<<(N_ATOM * ORIG + 255) / 256, 256, 0, stream>>>(atom_fea, mask, amf);

    // zero-padded conv weights (K 169 -> 172)
    pad_w_kernel<<<(NCONV * OC * KPAD + 255) / 256, 256, 0, stream>>>(conv_W, Wpad);

    // embedding: h0 = (atom_fea*mask) @ emb_W^T
    embed_kernel<<<(int)((N64 + 255) / 256), 256, 0, stream>>>(atom_fea, mask, emb_W, hA);

    float* hcur = hA;
    float* hnxt = hB;
    for (int layer = 0; layer < NCONV; ++layer) {
        zero_stats_kernel<<<1, 384, 0, stream>>>(stats1, stats2);
        conv_gemm_kernel<<<NM / 32, 256, 0, stream>>>(hcur, nbr_fea, nbr_idx,
                                                      Wpad + (size_t)layer * OC * KPAD,
                                                      conv_b + layer * OC,
                                                      gated, stats1);
        bn1_final_kernel<<<1, 128, 0, stream>>>(stats1, bn1_g + layer * OC,
                                                bn1_b + layer * OC, coef1);
        gate_reduce_kernel<<<N_ATOM, 64, 0, stream>>>(gated, coef1, summed);
        stats2_kernel<<<128, 256, 0, stream>>>(summed, stats2);
        bn2_final_kernel<<<1, 64, 0, stream>>>(stats2, bn2_g + layer * AF,
                                               bn2_b + layer * AF, coef2);
        update_h_kernel<<<(int)((N64 + 255) / 256), 256, 0, stream>>>(hcur, summed, coef2, hnxt);
        float* t = hcur; hcur = hnxt; hnxt = t;
    }

    readout_kernel<<<NCRY, 128, 0, stream>>>(hcur, cidx, fc1_W, fc1_b,
                                             fc2_W, fc2_b, out_W, out_b, props);
}